// OverAll_30442728194105
// MI455X (gfx1250) — compile-verified
//
#include <hip/hip_runtime.h>
#include <math.h>

#define EPSF 1e-12f
#define DD 128
#define OUTD 384
#define PXN 64
#define SOS 388   // padded LDS stride for 384-wide tiles (388 % 64 == 4)

typedef __attribute__((ext_vector_type(2))) float v2f;
typedef __attribute__((ext_vector_type(8))) float v8f;

// V_WMMA_F32_16X16X4_F32 : D(16x16,f32) = A(16x4,f32) x B(4x16,f32) + C
__device__ __forceinline__ v8f wmma4(v2f a, v2f b, v8f c) {
  return __builtin_amdgcn_wmma_f32_16x16x4_f32(false, a, false, b,
                                               (short)0, c, false, false);
}

__device__ __forceinline__ float wsum(float v) {
#pragma unroll
  for (int m = 16; m >= 1; m >>= 1) v += __shfl_xor(v, m, 32);
  return v;
}
__device__ __forceinline__ float wmaxr(float v) {
#pragma unroll
  for (int m = 16; m >= 1; m >>= 1) v = fmaxf(v, __shfl_xor(v, m, 32));
  return v;
}

// ---------------- row_softmax_matmul pieces ----------------
__global__ void k_deg(const int* __restrict__ mat, int nE, float* __restrict__ deg) {
  int i = blockIdx.x * blockDim.x + threadIdx.x;
  if (i < nE) atomicAdd(&deg[mat[2 * i]], 1.0f);
}

__global__ void k_scatter(const int* __restrict__ mat, int nE,
                          const float* __restrict__ deg,
                          const float* __restrict__ emb,
                          float* __restrict__ feat) {
  int t = blockIdx.x * blockDim.x + threadIdx.x;
  int i = t >> 5, l = t & 31;
  if (i >= nE) return;
  int row = mat[2 * i], col = mat[2 * i + 1];
  float w = 1.0f / fmaxf(deg[row], 1.0f);
#pragma unroll
  for (int j = 0; j < 4; ++j)
    atomicAdd(&feat[(long long)row * DD + l + 32 * j],
              w * emb[(long long)col * DD + l + 32 * j]);
}

// ---------------- per-relation stats: norm + attn dots ----------------
__global__ void k_relstats(const float* __restrict__ rel_emb, int nRel,
                           const float* __restrict__ eAttn,  // (2,128)
                           const float* __restrict__ rAttn,  // (2,128)
                           float* __restrict__ nrm, float* __restrict__ rdot) {
  int wave = (blockIdx.x * blockDim.x + threadIdx.x) >> 5;
  int l = threadIdx.x & 31;
  if (wave >= nRel) return;
  float ss = 0, d0 = 0, d1 = 0, d2 = 0, d3 = 0;
#pragma unroll
  for (int j = 0; j < 4; ++j) {
    float v = rel_emb[(long long)wave * DD + l + 32 * j];
    ss += v * v;
    d0 += v * eAttn[l + 32 * j];
    d1 += v * eAttn[DD + l + 32 * j];
    d2 += v * rAttn[l + 32 * j];
    d3 += v * rAttn[DD + l + 32 * j];
  }
  ss = wsum(ss); d0 = wsum(d0); d1 = wsum(d1); d2 = wsum(d2); d3 = wsum(d3);
  if (l == 0) {
    nrm[wave] = sqrtf(ss);
    rdot[4 * wave + 0] = d0; rdot[4 * wave + 1] = d1;
    rdot[4 * wave + 2] = d2; rdot[4 * wave + 3] = d3;
  }
}

// ---------------- per-edge scale + 4 logit arrays ----------------
__global__ void k_edgeprep(const int* __restrict__ rix, const float* __restrict__ rval,
                           int T, const float* __restrict__ nrm,
                           const float* __restrict__ rdot,
                           float* __restrict__ sv, float* __restrict__ lg) {
  int i = blockIdx.x * blockDim.x + threadIdx.x;
  if (i >= T) return;
  int k = rix[2 * i + 1];
  float rv = rval[i];
  float s = rv / fmaxf(rv * nrm[k], EPSF);
  sv[i] = s;
  lg[i]                     = s * rdot[4 * k + 0];
  lg[(long long)T + i]      = s * rdot[4 * k + 1];
  lg[2ll * T + i]           = s * rdot[4 * k + 2];
  lg[3ll * T + i]           = s * rdot[4 * k + 3];
}

// ---------------- CSR row ranges (adj sorted by src) ----------------
__global__ void k_rowptr(const int* __restrict__ adj, int T,
                         int* __restrict__ rs, int* __restrict__ re) {
  int i = blockIdx.x * blockDim.x + threadIdx.x;
  if (i >= T) return;
  int s = adj[2 * i];
  if (i == 0 || adj[2 * (i - 1)] != s) rs[s] = i;
  if (i == T - 1 || adj[2 * (i + 1)] != s) re[s] = i + 1;
}

// ---------------- feats0 = tanh(feature) into outputs cols [0,128) ------
__global__ void k_tanh_init(const float* __restrict__ feat, int n,
                            float* __restrict__ outBuf) {
  int t = blockIdx.x * blockDim.x + threadIdx.x;
  int node = t >> 5, l = t & 31;
  if (node >= n) return;
#pragma unroll
  for (int j = 0; j < 4; ++j)
    outBuf[(long long)node * OUTD + l + 32 * j] =
        tanhf(feat[(long long)node * DD + l + 32 * j]);
}

// ---------------- one attention depth: wave per node ----------------
__global__ void k_nodeagg(const int* __restrict__ adj, const int* __restrict__ rix,
                          const float* __restrict__ sv, const float* __restrict__ lg,
                          const int* __restrict__ rs, const int* __restrict__ re,
                          const float* __restrict__ rel_emb,
                          float* __restrict__ outBuf, int colIn, int colOut,
                          int nNode) {
  int wave = (blockIdx.x * blockDim.x + threadIdx.x) >> 5;
  int l = threadIdx.x & 31;
  if (wave >= nNode) return;
  int a = rs[wave], b = re[wave];
  float a0 = 0, a1 = 0, a2 = 0, a3 = 0;
  if (b > a) {
    float mx = -3.4e38f;
    for (int i = a + l; i < b; i += 32) mx = fmaxf(mx, lg[i]);
    mx = wmaxr(mx);
    float Z = 0;
    for (int i = a + l; i < b; i += 32) Z += expf(lg[i] - mx);
    Z = wsum(Z);
    float invZ = 1.0f / fmaxf(Z, EPSF);
    for (int i = a; i < b; ++i) {
      int d = adj[2 * i + 1];
      int k = rix[2 * i + 1];
      float s = sv[i];
      float alpha = expf(lg[i] - mx) * invZ;
      const float* vb = outBuf + (long long)d * OUTD + colIn;
      const float* rb = rel_emb + (long long)k * DD;
      float v0 = vb[l], v1 = vb[l + 32], v2 = vb[l + 64], v3 = vb[l + 96];
      float r0 = rb[l] * s, r1 = rb[l + 32] * s, r2 = rb[l + 64] * s, r3 = rb[l + 96] * s;
      float dot = wsum(v0 * r0 + v1 * r1 + v2 * r2 + v3 * r3);
      float c = 2.0f * dot;
      a0 += alpha * (v0 - c * r0);
      a1 += alpha * (v1 - c * r1);
      a2 += alpha * (v2 - c * r2);
      a3 += alpha * (v3 - c * r3);
    }
  }
  float* ob = outBuf + (long long)wave * OUTD + colOut;
  ob[l] = tanhf(a0); ob[l + 32] = tanhf(a1);
  ob[l + 64] = tanhf(a2); ob[l + 96] = tanhf(a3);
}

// ---------------- L2-normalized proxy rows ----------------
__global__ void k_pl2(const float* __restrict__ proxy, float* __restrict__ pl2,
                      int rows) {
  int wave = (blockIdx.x * blockDim.x + threadIdx.x) >> 5;
  int l = threadIdx.x & 31;
  if (wave >= rows) return;
  float ss = 0;
#pragma unroll
  for (int j = 0; j < 12; ++j) {
    float v = proxy[(long long)wave * OUTD + l + 32 * j];
    ss += v * v;
  }
  ss = wsum(ss);
  float inv = 1.0f / fmaxf(sqrtf(ss), EPSF);
#pragma unroll
  for (int j = 0; j < 12; ++j)
    pl2[(long long)wave * OUTD + l + 32 * j] =
        proxy[(long long)wave * OUTD + l + 32 * j] * inv;
}

// ------- fused proxy-attention + gate, WMMA f32, 16 rows per block -------
__global__ __launch_bounds__(256) void k_proxygate(
    const float* __restrict__ outBuf, const float* __restrict__ pl2,
    const float* __restrict__ proxy, const float* __restrict__ gateK,
    const float* __restrict__ bias, float* __restrict__ dOut, int outOff) {
  __shared__ float sOut[16 * SOS];
  __shared__ float sPf[16 * SOS];
  __shared__ float sAtt[16 * 68];
  __shared__ float sRn[16];
  const int r0 = blockIdx.x * 16;
  const int tid = threadIdx.x, wave = tid >> 5, l = tid & 31;
  const int am = l & 15, kb = (l >> 4) * 2;
  const int mb = (l >> 4) * 8, dn = l & 15;

  for (int t = tid; t < 16 * OUTD; t += 256) {
    int m = t / OUTD, c = t - m * OUTD;
    sOut[m * SOS + c] = outBuf[(long long)(r0 + m) * OUTD + c];
  }
  __syncthreads();

  // row inverse-norms
  for (int rr = wave; rr < 16; rr += 8) {
    float ss = 0;
#pragma unroll
    for (int j = 0; j < 12; ++j) { float v = sOut[rr * SOS + l + 32 * j]; ss += v * v; }
    ss = wsum(ss);
    if (l == 0) sRn[rr] = 1.0f / fmaxf(sqrtf(ss), EPSF);
  }
  __syncthreads();

  // logits = l2norm(out) @ pl2^T  -> sAtt (16x64), 4 N-tiles on waves 0..3
  if (wave < 4) {
    int n0 = wave * 16;
    v8f c = {0.f, 0.f, 0.f, 0.f, 0.f, 0.f, 0.f, 0.f};
    float rn = sRn[am];
    for (int k = 0; k < OUTD; k += 4) {
      v2f a, b;
      a.x = sOut[am * SOS + k + kb] * rn;
      a.y = sOut[am * SOS + k + kb + 1] * rn;
      b.x = pl2[(long long)(n0 + am) * OUTD + k + kb];      // B[k][n] = pl2[n][k]
      b.y = pl2[(long long)(n0 + am) * OUTD + k + kb + 1];
      c = wmma4(a, b, c);
    }
#pragma unroll
    for (int j = 0; j < 8; ++j) sAtt[(mb + j) * 68 + n0 + dn] = c[j];
  }
  __syncthreads();

  // row softmax over 64
  for (int rr = wave; rr < 16; rr += 8) {
    float v0 = sAtt[rr * 68 + l], v1 = sAtt[rr * 68 + 32 + l];
    float mx = wmaxr(fmaxf(v0, v1));
    float e0 = expf(v0 - mx), e1 = expf(v1 - mx);
    float Z = wsum(e0 + e1);
    float inv = 1.0f / fmaxf(Z, EPSF);
    sAtt[rr * 68 + l] = e0 * inv;
    sAtt[rr * 68 + 32 + l] = e1 * inv;
  }
  __syncthreads();

  // pf = out - att @ proxy : 24 N-tiles, 3 per wave, K = 64
  for (int tile = wave; tile < 24; tile += 8) {
    int c0 = tile * 16;
    v8f c = {0.f, 0.f, 0.f, 0.f, 0.f, 0.f, 0.f, 0.f};
    for (int k = 0; k < PXN; k += 4) {
      v2f a, b;
      a.x = sAtt[am * 68 + k + kb];
      a.y = sAtt[am * 68 + k + kb + 1];
      b.x = proxy[(long long)(k + kb) * OUTD + c0 + am];
      b.y = proxy[(long long)(k + kb + 1) * OUTD + c0 + am];
      c = wmma4(a, b, c);
    }
#pragma unroll
    for (int j = 0; j < 8; ++j) {
      int m = mb + j;
      sPf[m * SOS + c0 + dn] = sOut[m * SOS + c0 + dn] - c[j];
    }
  }
  __syncthreads();

  // gate = sigmoid(pf @ gateK + bias); out = g*out + (1-g)*pf
  for (int tile = wave; tile < 24; tile += 8) {
    int c0 = tile * 16;
    v8f c = {0.f, 0.f, 0.f, 0.f, 0.f, 0.f, 0.f, 0.f};
    for (int k = 0; k < OUTD; k += 4) {
      v2f a, b;
      a.x = sPf[am * SOS + k + kb];
      a.y = sPf[am * SOS + k + kb + 1];
      b.x = gateK[(long long)(k + kb) * OUTD + c0 + am];
      b.y = gateK[(long long)(k + kb + 1) * OUTD + c0 + am];
      c = wmma4(a, b, c);
    }
#pragma unroll
    for (int j = 0; j < 8; ++j) {
      int m = mb + j, cc = c0 + dn;
      float g = 1.0f / (1.0f + expf(-(c[j] + bias[cc])));
      float o = sOut[m * SOS + cc], p = sPf[m * SOS + cc];
      dOut[(long long)(r0 + m) * (2 * OUTD) + outOff + cc] = g * o + (1.0f - g) * p;
    }
  }
}

extern "C" void kernel_launch(void* const* d_in, const int* in_sizes, int n_in,
                              void* d_out, int out_size, void* d_ws, size_t ws_size,
                              hipStream_t stream) {
  const float* ent_emb = (const float*)d_in[0];
  const float* rel_emb = (const float*)d_in[1];
  const float* e_attn  = (const float*)d_in[2];
  const float* e_proxy = (const float*)d_in[3];
  const float* e_gate  = (const float*)d_in[4];
  const float* e_bias  = (const float*)d_in[5];
  const float* r_attn  = (const float*)d_in[6];
  const float* r_proxy = (const float*)d_in[7];
  const float* r_gate  = (const float*)d_in[8];
  const float* r_bias  = (const float*)d_in[9];
  const float* r_val   = (const float*)d_in[10];
  const int* ent_mat   = (const int*)d_in[11];
  const int* rel_mat   = (const int*)d_in[12];
  const int* adj       = (const int*)d_in[13];
  const int* r_index   = (const int*)d_in[14];

  const int N   = in_sizes[0] / DD;      // 50000 nodes
  const int R   = in_sizes[1] / DD;      // 2000 relations
  const int nEe = in_sizes[11] / 2;      // 400000
  const int nEr = in_sizes[12] / 2;      // 300000
  const int T   = in_sizes[13] / 2;      // 500000 triples

  char* base = (char*)d_ws;
  size_t off = 0;
  auto alloc = [&](size_t bytes) {
    size_t o = off; off += (bytes + 255) & ~(size_t)255; return o;
  };
  float* featE = (float*)(base + alloc((size_t)N * DD * 4));
  float* featR = (float*)(base + alloc((size_t)N * DD * 4));
  float* degE  = (float*)(base + alloc((size_t)N * 4));
  float* degR  = (float*)(base + alloc((size_t)N * 4));
  int*   rowS  = (int*)  (base + alloc((size_t)N * 4));
  int*   rowE  = (int*)  (base + alloc((size_t)N * 4));
  size_t zbytes = off;                       // zero everything above
  float* nrm   = (float*)(base + alloc((size_t)R * 4));
  float* rdot  = (float*)(base + alloc((size_t)R * 16));
  float* sv    = (float*)(base + alloc((size_t)T * 4));
  float* lg    = (float*)(base + alloc((size_t)T * 4 * 4));
  float* outE  = (float*)(base + alloc((size_t)N * OUTD * 4));
  float* outR  = (float*)(base + alloc((size_t)N * OUTD * 4));
  float* pl2E  = (float*)(base + alloc((size_t)PXN * OUTD * 4));
  float* pl2R  = (float*)(base + alloc((size_t)PXN * OUTD * 4));
  (void)ws_size; (void)n_in; (void)out_size;

  hipMemsetAsync(d_ws, 0, zbytes, stream);

  const int B = 256;
  // degree + mean-scatter (row_softmax_matmul)
  k_deg<<<(nEe + B - 1) / B, B, 0, stream>>>(ent_mat, nEe, degE);
  k_deg<<<(nEr + B - 1) / B, B, 0, stream>>>(rel_mat, nEr, degR);
  k_scatter<<<(nEe * 32 + B - 1) / B, B, 0, stream>>>(ent_mat, nEe, degE, ent_emb, featE);
  k_scatter<<<(nEr * 32 + B - 1) / B, B, 0, stream>>>(rel_mat, nEr, degR, rel_emb, featR);

  // relation stats, edge scalars/logits, CSR row ranges
  k_relstats<<<(R * 32 + B - 1) / B, B, 0, stream>>>(rel_emb, R, e_attn, r_attn, nrm, rdot);
  k_edgeprep<<<(T + B - 1) / B, B, 0, stream>>>(r_index, r_val, T, nrm, rdot, sv, lg);
  k_rowptr<<<(T + B - 1) / B, B, 0, stream>>>(adj, T, rowS, rowE);

  // depth-0 features
  k_tanh_init<<<(N * 32 + B - 1) / B, B, 0, stream>>>(featE, N, outE);
  k_tanh_init<<<(N * 32 + B - 1) / B, B, 0, stream>>>(featR, N, outR);

  // two attention depths per branch (wave per node)
  int aggB = (N * 32 + B - 1) / B;
  k_nodeagg<<<aggB, B, 0, stream>>>(adj, r_index, sv, lg + 0ll * T, rowS, rowE,
                                    rel_emb, outE, 0, 128, N);
  k_nodeagg<<<aggB, B, 0, stream>>>(adj, r_index, sv, lg + 1ll * T, rowS, rowE,
                                    rel_emb, outE, 128, 256, N);
  k_nodeagg<<<aggB, B, 0, stream>>>(adj, r_index, sv, lg + 2ll * T, rowS, rowE,
                                    rel_emb, outR, 0, 128, N);
  k_nodeagg<<<aggB, B, 0, stream>>>(adj, r_index, sv, lg + 3ll * T, rowS, rowE,
                                    rel_emb, outR, 128, 256, N);

  // normalized proxies
  k_pl2<<<(PXN * 32 + B - 1) / B, B, 0, stream>>>(e_proxy, pl2E, PXN);
  k_pl2<<<(PXN * 32 + B - 1) / B, B, 0, stream>>>(r_proxy, pl2R, PXN);

  // fused proxy attention + gating (WMMA), writes final halves of d_out
  int pgB = N / 16;
  k_proxygate<<<pgB, 256, 0, stream>>>(outE, pl2E, e_proxy, e_gate, e_bias,
                                       (float*)d_out, 0);
  k_proxygate<<<pgB, 256, 0, stream>>>(outR, pl2R, r_proxy, r_gate, r_bias,
                                       (float*)d_out, OUTD);
}